// ScaleGATEncoder_35150012351254
// MI455X (gfx1250) — compile-verified
//
#include <hip/hip_runtime.h>

typedef __attribute__((ext_vector_type(2))) float v2f;
typedef __attribute__((ext_vector_type(8))) float v8f;

#define NND 64
#define DD  128
#define EE  1024
#define HH  4
#define CC  32
#define LLAYERS 2
#define KOPT 5
#define LN_EPS 1e-5f

// Order-preserving float->uint key for atomic max on LDS (ds_max_rtn_u32)
__device__ __forceinline__ unsigned fkey(float f) {
  unsigned u = __float_as_uint(f);
  return (u & 0x80000000u) ? ~u : (u | 0x80000000u);
}
__device__ __forceinline__ float fkey_dec(unsigned k) {
  return (k & 0x80000000u) ? __uint_as_float(k & 0x7fffffffu)
                           : __uint_as_float(~k);
}

__device__ __forceinline__ v8f wmma_f32(v2f a, v2f b, v8f c) {
  return __builtin_amdgcn_wmma_f32_16x16x4_f32(false, a, false, b, (short)0, c,
                                               false, false);
}

__global__ __launch_bounds__(256)
void scalegat_fused_kernel(const int* __restrict__ x, const int* __restrict__ eidx,
                           const float* __restrict__ eattr, const float* __restrict__ attr_emb,
                           const float* __restrict__ opt_emb, const float* __restrict__ prior,
                           const float* __restrict__ Wl, const float* __restrict__ bl,
                           const float* __restrict__ Wr, const float* __restrict__ br,
                           const float* __restrict__ We, const float* __restrict__ att,
                           const float* __restrict__ conv_bias, const float* __restrict__ ln_g,
                           const float* __restrict__ ln_b, const float* __restrict__ W1,
                           const float* __restrict__ b1, const float* __restrict__ W2,
                           const float* __restrict__ b2,
                           float* __restrict__ f_out, float* __restrict__ rel_out,
                           float* __restrict__ hat_out)
{
  __shared__ float sh_h  [NND * DD];   // 32 KB  residual stream
  __shared__ float sh_xl [NND * DD];   // 32 KB  h @ Wl + bl
  __shared__ float sh_xr [NND * DD];   // 32 KB  h @ Wr + br
  __shared__ float sh_out[NND * DD];   // 32 KB  message accumulator
  __shared__ float sh_ex [EE * HH];    // 16 KB  alpha, then exp(alpha-mx)
  __shared__ float sh_ea [EE];         //  4 KB  edge_attr
  __shared__ float sh_We [DD];         // We[l] row
  __shared__ float sh_att[HH * CC];    // att[l]
  __shared__ unsigned sh_mx[NND * HH]; // per-(node,head) max keys
  __shared__ float sh_den[NND * HH];   // softmax denominators
  __shared__ float sh_rel[NND];

  const int b    = blockIdx.x;
  const int tid  = threadIdx.x;
  const int lane = tid & 31;   // wave32
  const int wave = tid >> 5;   // 8 waves
  const int* src = eidx;
  const int* dst = eidx + EE;
  float* hatb = hat_out + (size_t)b * NND * 2 * DD;

  // WMMA fragment geometry (ISA 7.12.2, 32-bit operands, wave32):
  //  A 16x4 : lane half selects K+0/1 vs K+2/3, VGPR selects K lsb, M = lane&15
  //  B 4x16 : same K mapping, N = lane&15
  //  C 16x16: VGPR r -> M = r + 8*(lane>>4), N = lane&15
  const int halfk = (lane >> 4) * 2;
  const int lcol  = lane & 15;

  // ---------------- Phase 0: T0 = (attr_emb + opt_emb[n, x[b,n]]) * prior ----
  for (int i = tid; i < NND * DD; i += 256) {
    int n = i >> 7, d = i & 127;
    int opt = x[b * NND + n];
    float t0 = (attr_emb[i] + opt_emb[(n * KOPT + opt) * DD + d]) * prior[n];
    sh_h[i] = t0;
    hatb[n * 2 * DD + d] = t0;          // hat_T[:, :D] = T0 (read back in MLP)
  }
  for (int i = tid; i < EE; i += 256) sh_ea[i] = eattr[i];
  __threadfence();                      // make T0 globally visible for MLP reads
  __syncthreads();

  // ---------------- GAT layers ----------------------------------------------
  for (int l = 0; l < LLAYERS; ++l) {
    for (int i = tid; i < DD; i += 256) {
      sh_We[i]  = We[l * DD + i];
      sh_att[i] = att[l * DD + i];
    }
    for (int i = tid; i < NND * HH; i += 256) {
      sh_mx[i]  = fkey(-__builtin_inff());
      sh_den[i] = 0.f;
    }
    for (int i = tid; i < NND * DD; i += 256) sh_out[i] = 0.f;
    __syncthreads();

    // --- xl = h@Wl + bl, xr = h@Wr + br via V_WMMA_F32_16X16X4_F32 ----------
    // 16 column groups (2 weight mats x 8 N-tiles), 2 per wave.
    // Register-block 4 M-tiles per group: each B fragment feeds 4 WMMAs.
    {
      const float* Wlp = Wl + l * DD * DD;
      const float* Wrp = Wr + l * DD * DD;
      const float* blp = bl + l * DD;
      const float* brp = br + l * DD;
      for (int g = wave * 2; g < wave * 2 + 2; ++g) {
        int which = g >> 3;                  // 0 -> Wl/xl, 1 -> Wr/xr
        int n0 = (g & 7) * 16;
        const float* Wp = which ? Wrp : Wlp;
        int bcol = n0 + lcol;
        v8f acc0 = {}, acc1 = {}, acc2 = {}, acc3 = {};
        const int abase = lcol * DD;
        for (int k0 = 0; k0 < DD; k0 += 4) {
          int k = k0 + halfk;
          v2f bb;
          bb.x = Wp[k * DD + bcol];
          bb.y = Wp[(k + 1) * DD + bcol];
          v2f a0, a1, a2, a3;
          a0.x = sh_h[abase + k];                a0.y = sh_h[abase + k + 1];
          a1.x = sh_h[abase + 16 * DD + k];      a1.y = sh_h[abase + 16 * DD + k + 1];
          a2.x = sh_h[abase + 32 * DD + k];      a2.y = sh_h[abase + 32 * DD + k + 1];
          a3.x = sh_h[abase + 48 * DD + k];      a3.y = sh_h[abase + 48 * DD + k + 1];
          acc0 = wmma_f32(a0, bb, acc0);
          acc1 = wmma_f32(a1, bb, acc1);
          acc2 = wmma_f32(a2, bb, acc2);
          acc3 = wmma_f32(a3, bb, acc3);
        }
        float bias = which ? brp[bcol] : blp[bcol];
        float* outp = which ? sh_xr : sh_xl;
        int crow = (lane >> 4) * 8;
        for (int r = 0; r < 8; ++r) {
          outp[(crow + r) * DD + bcol]              = acc0[r] + bias;
          outp[(16 + crow + r) * DD + bcol]         = acc1[r] + bias;
          outp[(32 + crow + r) * DD + bcol]         = acc2[r] + bias;
          outp[(48 + crow + r) * DD + bcol]         = acc3[r] + bias;
        }
      }
    }
    __syncthreads();

    // --- Edge pass 1: alpha = sum_c lrelu(msg+xr+e)*att ; scatter-max -------
    for (int ed = tid; ed < EE; ed += 256) {
      int s = src[ed], dn = dst[ed];
      float ea = sh_ea[ed];
      float al[HH] = {0.f, 0.f, 0.f, 0.f};
      const float* xlp = sh_xl + s  * DD;
      const float* xrp = sh_xr + dn * DD;
      for (int c = 0; c < DD; ++c) {
        float z = xlp[c] + xrp[c] + ea * sh_We[c];
        z = z > 0.f ? z : 0.2f * z;          // leaky_relu 0.2
        al[c >> 5] += z * sh_att[c];
      }
      for (int hh = 0; hh < HH; ++hh) {
        sh_ex[ed * HH + hh] = al[hh];
        atomicMax(&sh_mx[dn * HH + hh], fkey(al[hh]));
      }
    }
    __syncthreads();

    // --- Edge pass 2: ex = exp(alpha - mx[dst]); scatter-add denominator ----
    for (int ed = tid; ed < EE; ed += 256) {
      int dn = dst[ed];
      for (int hh = 0; hh < HH; ++hh) {
        float ex = __expf(sh_ex[ed * HH + hh] - fkey_dec(sh_mx[dn * HH + hh]));
        sh_ex[ed * HH + hh] = ex;
        atomicAdd(&sh_den[dn * HH + hh], ex);
      }
    }
    __syncthreads();

    // --- Edge pass 3: out[dst] += msg * (ex / den[dst]) ---------------------
    for (int ed = tid; ed < EE; ed += 256) {
      int s = src[ed], dn = dst[ed];
      float a4[HH];
      for (int hh = 0; hh < HH; ++hh)
        a4[hh] = sh_ex[ed * HH + hh] / sh_den[dn * HH + hh];
      const float* xlp = sh_xl + s * DD;
      float* op = sh_out + dn * DD;
      for (int c = 0; c < DD; ++c)
        atomicAdd(&op[c], xlp[c] * a4[c >> 5]);
    }
    __syncthreads();

    // --- conv_bias + LayerNorm + ELU + residual (one thread per node) -------
    if (tid < NND) {
      const float* cb = conv_bias + l * DD;
      const float* gg = ln_g + l * DD;
      const float* be = ln_b + l * DD;
      float* op = sh_out + tid * DD;
      float* hp = sh_h   + tid * DD;
      float mu = 0.f, m2 = 0.f;
      for (int c = 0; c < DD; ++c) {
        float v = op[c] + cb[c];
        mu += v; m2 += v * v;
      }
      mu *= (1.f / DD); m2 *= (1.f / DD);
      float rstd = rsqrtf(m2 - mu * mu + LN_EPS);
      for (int c = 0; c < DD; ++c) {
        float v = (op[c] + cb[c] - mu) * rstd * gg[c] + be[c];
        v = v > 0.f ? v : (__expf(v) - 1.f);     // ELU
        hp[c] += v;
      }
    }
    __syncthreads();
  }

  // ---------------- hat_T[:, D:] = h -----------------------------------------
  for (int i = tid; i < NND * DD; i += 256) {
    int n = i >> 7, d = i & 127;
    hatb[n * 2 * DD + DD + d] = sh_h[i];
  }
  __syncthreads();

  // ---------------- MLP: relu(hat_T(64x256) @ W1(256x128) + b1) -> sh_xl ----
  // 8 column groups x 4 M-tiles; one group per wave, B fragment reused 4x.
  {
    int n0 = wave * 16;
    int bcol = n0 + lcol;
    v8f acc0 = {}, acc1 = {}, acc2 = {}, acc3 = {};
    const int hrow = lcol * 2 * DD;          // hat_T row stride 256
    const int arow = lcol * DD;              // sh_h row stride 128
    for (int k0 = 0; k0 < 2 * DD; k0 += 4) {
      int k = k0 + halfk;
      v2f bb;
      bb.x = W1[k * DD + bcol];
      bb.y = W1[(k + 1) * DD + bcol];
      v2f a0, a1, a2, a3;
      if (k0 < DD) {                         // uniform branch: T0 half (global)
        a0.x = hatb[hrow + k];                    a0.y = hatb[hrow + k + 1];
        a1.x = hatb[hrow + 16 * 2 * DD + k];      a1.y = hatb[hrow + 16 * 2 * DD + k + 1];
        a2.x = hatb[hrow + 32 * 2 * DD + k];      a2.y = hatb[hrow + 32 * 2 * DD + k + 1];
        a3.x = hatb[hrow + 48 * 2 * DD + k];      a3.y = hatb[hrow + 48 * 2 * DD + k + 1];
      } else {                               // h half straight from LDS
        int kk = k - DD;
        a0.x = sh_h[arow + kk];               a0.y = sh_h[arow + kk + 1];
        a1.x = sh_h[arow + 16 * DD + kk];     a1.y = sh_h[arow + 16 * DD + kk + 1];
        a2.x = sh_h[arow + 32 * DD + kk];     a2.y = sh_h[arow + 32 * DD + kk + 1];
        a3.x = sh_h[arow + 48 * DD + kk];     a3.y = sh_h[arow + 48 * DD + kk + 1];
      }
      acc0 = wmma_f32(a0, bb, acc0);
      acc1 = wmma_f32(a1, bb, acc1);
      acc2 = wmma_f32(a2, bb, acc2);
      acc3 = wmma_f32(a3, bb, acc3);
    }
    float bias = b1[bcol];
    int crow = (lane >> 4) * 8;
    for (int r = 0; r < 8; ++r) {
      float v0 = acc0[r] + bias;
      float v1 = acc1[r] + bias;
      float v2 = acc2[r] + bias;
      float v3 = acc3[r] + bias;
      sh_xl[(crow + r) * DD + bcol]      = v0 > 0.f ? v0 : 0.f;
      sh_xl[(16 + crow + r) * DD + bcol] = v1 > 0.f ? v1 : 0.f;
      sh_xl[(32 + crow + r) * DD + bcol] = v2 > 0.f ? v2 : 0.f;
      sh_xl[(48 + crow + r) * DD + bcol] = v3 > 0.f ? v3 : 0.f;
    }
  }
  __syncthreads();

  // ---------------- rel = sigmoid(relu_out @ W2 + b2) ------------------------
  if (tid < NND) {
    const float* rp = sh_xl + tid * DD;
    float s = 0.f;
    for (int c = 0; c < DD; ++c) s += rp[c] * W2[c];
    s += b2[0];
    float r = 1.f / (1.f + __expf(-s));
    sh_rel[tid] = r;
    rel_out[b * NND + tid] = r;
  }
  __syncthreads();

  // ---------------- f_scale = sum_n h*rel / (sum_n rel + 1e-8) ---------------
  if (tid < DD) {
    float num = 0.f, den = 0.f;
    for (int n = 0; n < NND; ++n) {
      float r = sh_rel[n];
      num += sh_h[n * DD + tid] * r;
      den += r;
    }
    f_out[b * DD + tid] = num / (den + 1e-8f);
  }
}

extern "C" void kernel_launch(void* const* d_in, const int* in_sizes, int n_in,
                              void* d_out, int out_size, void* d_ws, size_t ws_size,
                              hipStream_t stream) {
  (void)in_sizes; (void)n_in; (void)d_ws; (void)ws_size; (void)out_size;
  const int*   x        = (const int*)  d_in[0];
  const int*   eidx     = (const int*)  d_in[1];
  const float* eattr    = (const float*)d_in[2];
  const float* attr_emb = (const float*)d_in[3];
  const float* opt_emb  = (const float*)d_in[4];
  const float* prior    = (const float*)d_in[5];
  const float* Wl       = (const float*)d_in[6];
  const float* bl       = (const float*)d_in[7];
  const float* Wr       = (const float*)d_in[8];
  const float* br       = (const float*)d_in[9];
  const float* We       = (const float*)d_in[10];
  const float* att      = (const float*)d_in[11];
  const float* cb       = (const float*)d_in[12];
  const float* lg       = (const float*)d_in[13];
  const float* lb       = (const float*)d_in[14];
  const float* W1       = (const float*)d_in[15];
  const float* b1       = (const float*)d_in[16];
  const float* W2       = (const float*)d_in[17];
  const float* b2       = (const float*)d_in[18];

  float* f_out   = (float*)d_out;                 // (B, D)   = 131072
  float* rel_out = f_out + 1024 * 128;            // (B, N)   = 65536
  float* hat_out = rel_out + 1024 * 64;           // (B, N, 2D)

  scalegat_fused_kernel<<<1024, 256, 0, stream>>>(
      x, eidx, eattr, attr_emb, opt_emb, prior, Wl, bl, Wr, br, We, att,
      cb, lg, lb, W1, b1, W2, b2, f_out, rel_out, hat_out);
}